// MuleGNN_20615843021436
// MI455X (gfx1250) — compile-verified
//
#include <hip/hip_runtime.h>
#include <hip/hip_bf16.h>
#include <cstdint>

// GraphSAGE 3-layer GNN for MI455X (gfx1250, wave32).
// Phases: degree -> (zero, scatter-atomic, WMMA-f32 GEMM+bias+ReLU) x3 -> classifier.

typedef __attribute__((ext_vector_type(2))) float v2f;
typedef __attribute__((ext_vector_type(8))) float v8f;

#define GNN_N 100000
#define GNN_E 1600000

// ---------------- utility kernels ----------------

__global__ void gnn_fill_zero(float* __restrict__ p, size_t n) {
    size_t i = (size_t)blockIdx.x * blockDim.x + threadIdx.x;
    if (i < n) p[i] = 0.0f;
}

__global__ void gnn_degree(const long long* __restrict__ dst,
                           float* __restrict__ deg, int n_edges) {
    int e = blockIdx.x * blockDim.x + threadIdx.x;
    if (e < n_edges) atomicAdd(&deg[(int)dst[e]], 1.0f);
}

__global__ void gnn_invdeg(float* __restrict__ deg, int n) {
    int i = blockIdx.x * blockDim.x + threadIdx.x;
    if (i < n) deg[i] = 1.0f / fmaxf(deg[i], 1.0f);
}

// Pack row-major W[Nout x K] into WMMA-fragment order so the GEMM's B-loads
// are contiguous per wave:  P[((nt*(K/4)+ks)*32 + lane)*2 + j]
//   = W[nt*16 + (lane&15)][ks*4 + 2*(lane>>4) + j]
__global__ void gnn_pack_w(const float* __restrict__ W, float* __restrict__ P,
                           int Nout, int K) {
    int t = blockIdx.x * blockDim.x + threadIdx.x;
    int total = (Nout / 16) * (K / 4) * 32;
    if (t >= total) return;
    int lane = t & 31;
    int rest = t >> 5;
    int ks = rest % (K / 4);
    int nt = rest / (K / 4);
    int n = nt * 16 + (lane & 15);
    int k = ks * 4 + 2 * (lane >> 4);
    P[(size_t)t * 2 + 0] = W[(size_t)n * K + k];
    P[(size_t)t * 2 + 1] = W[(size_t)n * K + k + 1];
}

// One wave per edge: lanes load 4 contiguous floats of x[src] (512B coalesced)
// and atomically add into summed[dst] (512B coalesced atomics, L2-resident).
__global__ void gnn_scatter(const float* __restrict__ xin,
                            const long long* __restrict__ src,
                            const long long* __restrict__ dst,
                            float* __restrict__ summed, int n_edges) {
    int t = blockIdx.x * blockDim.x + threadIdx.x;
    int e = t >> 5;
    int lane = t & 31;
    if (e >= n_edges) return;
    int s = (int)src[e];
    int d = (int)dst[e];
    const float4 v = *(const float4*)(xin + (size_t)s * 128 + lane * 4);
    float* o = summed + (size_t)d * 128 + lane * 4;
    atomicAdd(o + 0, v.x);
    atomicAdd(o + 1, v.y);
    atomicAdd(o + 2, v.z);
    atomicAdd(o + 3, v.w);
}

// ---------------- fused SAGE GEMM: out = relu(agg*invdeg @ Wl^T + x @ Wr^T + b)
// Full-precision f32 WMMA (V_WMMA_F32_16X16X4_F32), K=128, Nout = NT*16.
// 2 waves per block, each wave owns one 16-row tile; 6250 tiles exactly cover
// 100000 rows -> EXEC always all-ones (WMMA requirement).
template <int NT>
__global__ __launch_bounds__(64) void gnn_sage_gemm(
    const float* __restrict__ summed, const float* __restrict__ xin,
    const float* __restrict__ inv_deg,
    const float* __restrict__ Pl, const float* __restrict__ Pr,
    const float* __restrict__ bias, float* __restrict__ out) {
    constexpr int K = 128;
    constexpr int KS = K / 4;
    constexpr int NOUT = NT * 16;
    constexpr int LDA = 132;  // padded stride: banks (4m + k) distinct per wave

    __shared__ float lds[2 * 16 * LDA];
    const int wave = threadIdx.x >> 5;
    const int lane = threadIdx.x & 31;
    const int tile = blockIdx.x * 2 + wave;
    const int r0 = tile * 16;
    const int m = lane & 15;
    const int kh = lane >> 4;
    float* tbuf = lds + wave * 16 * LDA;

    v8f acc[NT];
#pragma unroll
    for (int i = 0; i < NT; ++i)
#pragma unroll
        for (int j = 0; j < 8; ++j) acc[i][j] = 0.0f;

#pragma unroll 1
    for (int pass = 0; pass < 2; ++pass) {
        const float* srcm = pass ? xin : summed;
        const v2f* P = (const v2f*)(pass ? Pr : Pl);

        // Stage (and pre-scale) one 16x128 A-tile into padded LDS, coalesced.
#pragma unroll 4
        for (int r = 0; r < 16; ++r) {
            float sc = pass ? 1.0f : inv_deg[r0 + r];
            const float4 v = *(const float4*)(srcm + (size_t)(r0 + r) * K + lane * 4);
            float* d = tbuf + r * LDA + lane * 4;
            d[0] = v.x * sc;
            d[1] = v.y * sc;
            d[2] = v.z * sc;
            d[3] = v.w * sc;
        }
        __syncthreads();

#pragma unroll 4
        for (int ks = 0; ks < KS; ++ks) {
            const int k0 = ks * 4;
            // A frag: 16x4 f32, lane = M, kh selects K-half, 8B-aligned ds load
            const v2f a = *(const v2f*)(tbuf + m * LDA + k0 + 2 * kh);
            const v2f* bp = P + (size_t)ks * 32 + lane;
#pragma unroll
            for (int nt = 0; nt < NT; ++nt) {
                const v2f b = bp[(size_t)nt * KS * 32];
                acc[nt] = __builtin_amdgcn_wmma_f32_16x16x4_f32(
                    false, a, false, b, (short)0, acc[nt], false, false);
            }
        }
        __syncthreads();
    }

    // Epilogue: bias + ReLU + store. VGPR v -> row r0+v+8*kh, col nt*16+m.
#pragma unroll
    for (int nt = 0; nt < NT; ++nt) {
        const float bv = bias[nt * 16 + m];
#pragma unroll
        for (int v = 0; v < 8; ++v) {
            float val = acc[nt][v] + bv;
            val = fmaxf(val, 0.0f);
            out[(size_t)(r0 + v + 8 * kh) * NOUT + nt * 16 + m] = val;
        }
    }
}

// Final classifier: sigmoid(h3 @ Wc^T + bc), D_OUT = 64.
__global__ void gnn_classifier(const float* __restrict__ h3,
                               const float* __restrict__ Wc,
                               const float* __restrict__ bc,
                               float* __restrict__ out, int n) {
    int i = blockIdx.x * blockDim.x + threadIdx.x;
    if (i >= n) return;
    const float4* hp = (const float4*)(h3 + (size_t)i * 64);
    const float4* wp = (const float4*)Wc;
    float s = 0.0f;
#pragma unroll
    for (int j = 0; j < 16; ++j) {
        float4 a = hp[j], b = wp[j];
        s += a.x * b.x + a.y * b.y + a.z * b.z + a.w * b.w;
    }
    s += bc[0];
    out[i] = 1.0f / (1.0f + expf(-s));
}

// ---------------- launch ----------------

extern "C" void kernel_launch(void* const* d_in, const int* in_sizes, int n_in,
                              void* d_out, int out_size, void* d_ws, size_t ws_size,
                              hipStream_t stream) {
    (void)in_sizes; (void)n_in; (void)out_size; (void)ws_size;
    constexpr size_t N = GNN_N;
    constexpr size_t E = GNN_E;

    const float* x = (const float*)d_in[0];
    const long long* ei = (const long long*)d_in[1];
    const long long* src = ei;
    const long long* dst = ei + E;
    const float* Wl1 = (const float*)d_in[2];
    const float* bl1 = (const float*)d_in[3];
    const float* Wr1 = (const float*)d_in[4];
    const float* Wl2 = (const float*)d_in[5];
    const float* bl2 = (const float*)d_in[6];
    const float* Wr2 = (const float*)d_in[7];
    const float* Wl3 = (const float*)d_in[8];
    const float* bl3 = (const float*)d_in[9];
    const float* Wr3 = (const float*)d_in[10];
    const float* Wc = (const float*)d_in[11];
    const float* bc = (const float*)d_in[12];
    float* outp = (float*)d_out;

    float* ws = (float*)d_ws;
    float* invd = ws;                        // N
    float* summed = invd + N;                // N*128
    float* h1 = summed + N * 128;            // N*128
    float* h2 = h1 + N * 128;                // N*128
    float* h3 = h1;                          // reuse: h1 is dead by layer 3 GEMM
    float* pw = h2 + N * 128;
    float* Pl1 = pw;                         // 128*128 each
    float* Pr1 = Pl1 + 16384;
    float* Pl2 = Pr1 + 16384;
    float* Pr2 = Pl2 + 16384;
    float* Pl3 = Pr2 + 16384;                // 64*128 each
    float* Pr3 = Pl3 + 8192;

    const int B = 256;
    auto blocks = [](size_t work, int b) { return (unsigned)((work + b - 1) / b); };

    // Degrees -> inverse degrees (shared by all 3 layers).
    gnn_fill_zero<<<blocks(N, B), B, 0, stream>>>(invd, N);
    gnn_degree<<<blocks(E, B), B, 0, stream>>>(dst, invd, (int)E);
    gnn_invdeg<<<blocks(N, B), B, 0, stream>>>(invd, (int)N);

    // Pack weights into WMMA fragment order (coalesced hot-loop B loads).
    gnn_pack_w<<<blocks(8 * 32 * 32, B), B, 0, stream>>>(Wl1, Pl1, 128, 128);
    gnn_pack_w<<<blocks(8 * 32 * 32, B), B, 0, stream>>>(Wr1, Pr1, 128, 128);
    gnn_pack_w<<<blocks(8 * 32 * 32, B), B, 0, stream>>>(Wl2, Pl2, 128, 128);
    gnn_pack_w<<<blocks(8 * 32 * 32, B), B, 0, stream>>>(Wr2, Pr2, 128, 128);
    gnn_pack_w<<<blocks(4 * 32 * 32, B), B, 0, stream>>>(Wl3, Pl3, 64, 128);
    gnn_pack_w<<<blocks(4 * 32 * 32, B), B, 0, stream>>>(Wr3, Pr3, 64, 128);

    const unsigned gemm_blocks = (unsigned)(N / 16 / 2);  // 3125, exact
    const unsigned scat_blocks = blocks(E * 32, B);

    // Layer 1
    gnn_fill_zero<<<blocks(N * 128, B), B, 0, stream>>>(summed, N * 128);
    gnn_scatter<<<scat_blocks, B, 0, stream>>>(x, src, dst, summed, (int)E);
    gnn_sage_gemm<8><<<gemm_blocks, 64, 0, stream>>>(summed, x, invd, Pl1, Pr1, bl1, h1);

    // Layer 2
    gnn_fill_zero<<<blocks(N * 128, B), B, 0, stream>>>(summed, N * 128);
    gnn_scatter<<<scat_blocks, B, 0, stream>>>(h1, src, dst, summed, (int)E);
    gnn_sage_gemm<8><<<gemm_blocks, 64, 0, stream>>>(summed, h1, invd, Pl2, Pr2, bl2, h2);

    // Layer 3 (Nout = 64)
    gnn_fill_zero<<<blocks(N * 128, B), B, 0, stream>>>(summed, N * 128);
    gnn_scatter<<<scat_blocks, B, 0, stream>>>(h2, src, dst, summed, (int)E);
    gnn_sage_gemm<4><<<gemm_blocks, 64, 0, stream>>>(summed, h2, invd, Pl3, Pr3, bl3, h3);

    // Classifier
    gnn_classifier<<<blocks(N, B), B, 0, stream>>>(h3, Wc, bc, outp, (int)N);
}